// SimpleGNN_72267119722864
// MI455X (gfx1250) — compile-verified
//
#include <hip/hip_runtime.h>
#include <hip/hip_bf16.h>

typedef float v2f __attribute__((ext_vector_type(2)));
typedef float v8f __attribute__((ext_vector_type(8)));

#define K_DIM 64

__device__ __forceinline__ void atomAddF(float* p, float v) {
    __hip_atomic_fetch_add(p, v, __ATOMIC_RELAXED, __HIP_MEMORY_SCOPE_AGENT);
}

// ---------------------------------------------------------------------------
// Degree / normalization kernels
// ---------------------------------------------------------------------------
__global__ void k_deg_init(float* deg, int n) {
    int i = blockIdx.x * blockDim.x + threadIdx.x;
    if (i < n) deg[i] = 1.0f;  // self-loop contributes 1 to every node's degree
}

__global__ void k_deg_accum(const long long* __restrict__ ei, float* deg, int E) {
    int e = blockIdx.x * blockDim.x + threadIdx.x;
    if (e < E) {
        int d = (int)ei[E + e];  // dst row of edge_index
        atomAddF(&deg[d], 1.0f);
    }
}

__global__ void k_deg_rsqrt(float* deg, int n) {
    int i = blockIdx.x * blockDim.x + threadIdx.x;
    if (i < n) deg[i] = rsqrtf(deg[i]);  // deg >= 1 always (self-loop)
}

// ---------------------------------------------------------------------------
// Pad W2 [64x10] -> W2p [64x16] with zero columns (one-shot tiny kernel)
// ---------------------------------------------------------------------------
__global__ void k_pad_w2(const float* __restrict__ W2, float* __restrict__ W2p) {
    int i = blockIdx.x * blockDim.x + threadIdx.x;  // 64*16 threads
    if (i >= 64 * 16) return;
    int k = i >> 4;
    int n = i & 15;
    W2p[i] = (n < 10) ? W2[k * 10 + n] : 0.0f;
}

// ---------------------------------------------------------------------------
// fp32 WMMA GEMM:  C[M x NB] = A[M x 64] * B[64 x NB], NB compile-time.
// One wave computes one 16x16 output tile via 16 chained V_WMMA_F32_16X16X4_F32.
// A 16x4 layout: lane<16 -> M=lane; vgpr j holds K = j + 2*(lane>=16).
// B 4x16 / C 16x16: rows striped across lanes within a VGPR per ISA layout.
// No masking anywhere: EXEC stays all-ones per WMMA requirements.
// ---------------------------------------------------------------------------
template <int NB>
__global__ void k_gemm_wmma_f32(const float* __restrict__ A,
                                const float* __restrict__ B,
                                float* __restrict__ C,
                                int nTiles) {
    constexpr int nTilesN = NB / 16;
    int wave = (int)((blockIdx.x * blockDim.x + threadIdx.x) >> 5);
    if (wave >= nTiles) return;
    int lane = threadIdx.x & 31;
    int tLo  = lane & 15;
    int tHi  = lane >> 4;

    int tm = wave / nTilesN;
    int tn = wave - tm * nTilesN;

    const float* Arow = A + (size_t)(tm * 16 + tLo) * K_DIM;
    const float* Bcol = B + tn * 16 + tLo;

    v8f acc = {};
#pragma unroll
    for (int kb = 0; kb < K_DIM; kb += 4) {
        int k0 = kb + 2 * tHi;               // K of vgpr 0; vgpr 1 is k0+1
        v2f a = *(const v2f*)(Arow + k0);    // 8B-aligned pair -> global_load_b64
        v2f b;
        b.x = Bcol[(size_t)k0 * NB];
        b.y = Bcol[(size_t)(k0 + 1) * NB];
        acc = __builtin_amdgcn_wmma_f32_16x16x4_f32(
            /*neg_a=*/false, a, /*neg_b=*/false, b,
            /*c_mod=*/(short)0, acc, /*reuse_a=*/false, /*reuse_b=*/false);
    }

    float* Crow = C + (size_t)tm * 16 * NB + tn * 16 + tLo;
#pragma unroll
    for (int r = 0; r < 8; ++r) {
        // C/D layout: VGPR r, lanes 0-15 -> M=r; lanes 16-31 -> M=r+8
        Crow[(size_t)(r + 8 * tHi) * NB] = acc[r];
    }
}

// ---------------------------------------------------------------------------
// y1[i][f] = h[i][f] * dis[i]^2 + b1[f]   (self-loop message + bias), F=64
// ---------------------------------------------------------------------------
__global__ void k_layer_init64(const float* __restrict__ h,
                               const float* __restrict__ dis,
                               const float* __restrict__ bias,
                               float* __restrict__ out, int n) {
    int i = blockIdx.x * blockDim.x + threadIdx.x;
    if (i >= n * 64) return;
    int node = i >> 6;
    int f    = i & 63;
    float di = dis[node];
    out[i] = h[i] * di * di + bias[f];
}

// ---------------------------------------------------------------------------
// out[i][c] = h2p[i][c] * dis[i]^2 + b2[c]; h2p padded stride 16, out stride 10
// ---------------------------------------------------------------------------
__global__ void k_out_init(const float* __restrict__ h2p,
                           const float* __restrict__ dis,
                           const float* __restrict__ b2,
                           float* __restrict__ out, int n) {
    int i = blockIdx.x * blockDim.x + threadIdx.x;
    if (i >= n * 10) return;
    int node = i / 10;
    int c    = i - node * 10;
    float di = dis[node];
    out[i] = h2p[node * 16 + c] * di * di + b2[c];
}

// ---------------------------------------------------------------------------
// Edge aggregation, 64 features: one wave per edge, lane covers f and f+32.
// Gather h[src] (coalesced 128B lines, L2-resident) and atomically scatter
// into out[dst]. norm = dis[src]*dis[dst].
// ---------------------------------------------------------------------------
__global__ void k_edge_agg64(const long long* __restrict__ ei,
                             const float* __restrict__ h,
                             const float* __restrict__ dis,
                             float* __restrict__ out, int E) {
    unsigned gid = blockIdx.x * blockDim.x + threadIdx.x;
    int e = (int)(gid >> 5);
    if (e >= E) return;
    int lane = threadIdx.x & 31;
    int s = (int)ei[e];
    int d = (int)ei[E + e];
    float w = dis[s] * dis[d];
    float v0 = h[(size_t)s * 64 + lane];
    float v1 = h[(size_t)s * 64 + lane + 32];
    atomAddF(&out[(size_t)d * 64 + lane],      v0 * w);
    atomAddF(&out[(size_t)d * 64 + lane + 32], v1 * w);
}

// ---------------------------------------------------------------------------
// Edge aggregation, 10 features (h2p padded stride 16): 16 lanes per edge.
// ---------------------------------------------------------------------------
__global__ void k_edge_agg10(const long long* __restrict__ ei,
                             const float* __restrict__ h2p,
                             const float* __restrict__ dis,
                             float* __restrict__ out, int E) {
    unsigned gid = blockIdx.x * blockDim.x + threadIdx.x;
    int e = (int)(gid >> 4);
    if (e >= E) return;
    int lane = threadIdx.x & 15;
    if (lane >= 10) return;
    int s = (int)ei[e];
    int d = (int)ei[E + e];
    float w = dis[s] * dis[d];
    atomAddF(&out[(size_t)d * 10 + lane], h2p[(size_t)s * 16 + lane] * w);
}

// ---------------------------------------------------------------------------
extern "C" void kernel_launch(void* const* d_in, const int* in_sizes, int n_in,
                              void* d_out, int out_size, void* d_ws, size_t ws_size,
                              hipStream_t stream) {
    const float*     x  = (const float*)d_in[0];      // [N,64]
    const float*     W1 = (const float*)d_in[1];      // [64,64]
    const float*     b1 = (const float*)d_in[2];      // [64]
    const float*     W2 = (const float*)d_in[3];      // [64,10]
    const float*     b2 = (const float*)d_in[4];      // [10]
    const long long* ei = (const long long*)d_in[5];  // [2,E] int64

    const int N = in_sizes[0] / 64;   // 100000
    const int E = in_sizes[5] / 2;    // 1600000

    float* ws  = (float*)d_ws;
    float* h   = ws;                                   // [N,64] projected features
    float* y1  = ws + (size_t)N * 64;                  // [N,64] layer-1 output
    float* h2p = ws + (size_t)N * 128;                 // [N,16] padded layer-2 proj
    float* dis = ws + (size_t)N * 144;                 // [N] deg^{-1/2}
    float* W2p = ws + (size_t)N * 145;                 // [64,16] padded W2
    float* out = (float*)d_out;                        // [N,10]

    const int TB = 256;

    // 1) degrees with self-loop, then deg^{-1/2}; pad W2 in parallel
    k_deg_init<<<(N + TB - 1) / TB, TB, 0, stream>>>(dis, N);
    k_pad_w2<<<4, TB, 0, stream>>>(W2, W2p);
    k_deg_accum<<<(E + TB - 1) / TB, TB, 0, stream>>>(ei, dis, E);
    k_deg_rsqrt<<<(N + TB - 1) / TB, TB, 0, stream>>>(dis, N);

    // 2) h = x @ W1   (WMMA fp32, 6250x4 tiles, 8 waves/block)
    {
        int nTiles = (N / 16) * 4;           // 25000
        int blocks = (nTiles + 7) / 8;
        k_gemm_wmma_f32<64><<<blocks, TB, 0, stream>>>(x, W1, h, nTiles);
    }

    // 3) y1 = h*dis^2 + b1, then edge scatter-add into y1
    k_layer_init64<<<((size_t)N * 64 + TB - 1) / TB, TB, 0, stream>>>(h, dis, b1, y1, N);
    {
        unsigned long long thr = (unsigned long long)E * 32ull;
        k_edge_agg64<<<(unsigned)((thr + TB - 1) / TB), TB, 0, stream>>>(ei, h, dis, y1, E);
    }

    // 4) h2p = y1 @ W2p   (WMMA fp32, one unmasked padded N-tile per 16 rows)
    {
        int nTiles = N / 16;                 // 6250
        int blocks = (nTiles + 7) / 8;
        k_gemm_wmma_f32<16><<<blocks, TB, 0, stream>>>(y1, W2p, h2p, nTiles);
    }

    // 5) out = h2p*dis^2 + b2, then edge scatter-add into out
    k_out_init<<<((size_t)N * 10 + TB - 1) / TB, TB, 0, stream>>>(h2p, dis, b2, out, N);
    {
        unsigned long long thr = (unsigned long long)E * 16ull;
        k_edge_agg10<<<(unsigned)((thr + TB - 1) / TB), TB, 0, stream>>>(ei, h2p, dis, out, E);
    }
}